// CausalAttention_75350906241536
// MI455X (gfx1250) — compile-verified
//
#include <hip/hip_runtime.h>
#include <hip/hip_bf16.h>

typedef float v2f __attribute__((ext_vector_type(2)));
typedef float v8f __attribute__((ext_vector_type(8)));

constexpr int BB = 2, TT = 512, HH = 256;
constexpr int KST = HH + 4;          // padded LDS row stride (floats): kills bank conflicts

// tanh(x) = 1 - 2/(exp(2x)+1), via v_exp_f32 (exp2) + v_rcp_f32.
// Saturates correctly: x>>0 -> exp2->inf -> rcp->0 -> 1 ; x<<0 -> 1-2 = -1.
__device__ __forceinline__ float fast_tanh(float x) {
    float e = __builtin_amdgcn_exp2f(x * 2.8853900817779268f); // 2*log2(e)
    return 1.0f - 2.0f * __builtin_amdgcn_rcpf(e + 1.0f);
}

// Issue one 64-row Kp chunk (64 x 256 floats) as async global->LDS B128 copies.
// Each of 256 threads issues 16 x 16B loads; coalesced in global, padded rows in LDS.
// Tracked by ASYNCcnt; caller waits with s_wait_asynccnt + barrier.
__device__ __forceinline__ void issue_chunk(const float* __restrict__ gbase,
                                            float* lbuf, int tid)
{
    uint32_t lds_base = (uint32_t)(uintptr_t)lbuf;   // low 32 bits = LDS byte offset
#pragma unroll
    for (int it = 0; it < 16; ++it) {
        int idx = it * 256 + tid;     // float4 index within chunk (0..4095)
        int row = idx >> 6;           // 64 float4s per 256-float row
        int c4  = idx & 63;
        uint32_t laddr = lds_base + (uint32_t)(row * KST + c4 * 4) * 4u;
        const float* g = gbase + (size_t)row * HH + c4 * 4;
        asm volatile("global_load_async_to_lds_b128 %0, %1, off"
                     :: "v"(laddr), "v"(g) : "memory");
    }
}

// ---------------- Kernel A: Qp = Q*Wq^T, Kp = K*Wk^T via V_WMMA_F32_16X16X4_F32
// One wave per 16x16 output tile. 2048 waves total (2 proj * 2 batch * 32 * 16).
__global__ __launch_bounds__(256) void proj_kernel(
    const float* __restrict__ Q, const float* __restrict__ K,
    const float* __restrict__ Wq, const float* __restrict__ Wk,
    float* __restrict__ Qp, float* __restrict__ Kp)
{
    int wave = threadIdx.x >> 5;
    int lane = threadIdx.x & 31;
    int tile = blockIdx.x * 8 + wave;       // 0..2047
    int which = tile >> 10;                 // 0 => Q-proj, 1 => K-proj
    int r  = tile & 1023;
    int b  = r >> 9;                        // batch
    int rt = (r & 511) >> 4;                // row tile (T/16 = 32)
    int ct = r & 15;                        // col tile (H/16 = 16)

    const float* In = which ? K  : Q;
    const float* W  = which ? Wk : Wq;
    float*       Out = which ? Kp : Qp;

    int m  = lane & 15;                     // A: row M / B: col N (striped over lanes)
    int kk = (lane >> 4) * 2;               // K sub-offset per half-wave

    const float* inRow = In + (size_t)b * TT * HH + (size_t)(rt * 16 + m) * HH;
    const float* wRow  = W  + (size_t)(ct * 16 + m) * HH;   // B[k,n] = W[n,k]

    v8f c = {};
    for (int k = 0; k < HH; k += 4) {
        v2f a, bm;
        a.x  = inRow[k + kk];     a.y  = inRow[k + kk + 1];
        bm.x = wRow[k + kk];      bm.y = wRow[k + kk + 1];
        c = __builtin_amdgcn_wmma_f32_16x16x4_f32(false, a, false, bm,
                                                  (short)0, c, false, false);
    }
    // C/D layout: VGPR i holds M = i + 8*(lane>>4), N = lane&15
    float* outBase = Out + (size_t)b * TT * HH
                   + (size_t)(rt * 16 + 8 * (lane >> 4)) * HH + ct * 16 + m;
#pragma unroll
    for (int i = 0; i < 8; ++i) outBase[(size_t)i * HH] = c[i];
}

// ---------------- Kernel B: fused additive scores + causal softmax, one WG per (b,t)
// Double-buffered async Kp chunks (ASYNCcnt) overlap global traffic with tanh VALU work.
__global__ __launch_bounds__(256) void score_softmax_kernel(
    const float* __restrict__ Qp, const float* __restrict__ Kp,
    const float* __restrict__ v, float* __restrict__ alpha)
{
    __shared__ float qp_s[HH];
    __shared__ float vv_s[HH];
    __shared__ float ks[2][64 * KST];   // 2 x 64 padded Kp rows (~130 KB)
    __shared__ float srow[TT];
    __shared__ float red[256];

    int tid = threadIdx.x;
    int t   = blockIdx.x;
    int b   = blockIdx.y;

    qp_s[tid] = Qp[(size_t)b * TT * HH + (size_t)t * HH + tid];
    vv_s[tid] = v[tid];

    const float* kb = Kp + (size_t)b * TT * HH;

    // Prologue: async-load chunk 0 into buffer 0.
    issue_chunk(kb, &ks[0][0], tid);
    asm volatile("s_wait_asynccnt 0x0" ::: "memory");
    __syncthreads();

    int sl   = tid >> 2;   // key index within chunk (0..63)
    int part = tid & 3;    // 4 adjacent lanes per key, interleaved float4 partitions

    for (int s0 = 0; s0 <= t; s0 += 64) {
        int buf = (s0 >> 6) & 1;
        // Prefetch next chunk into the other buffer while we compute this one.
        if (s0 + 64 <= t) issue_chunk(kb + (size_t)(s0 + 64) * HH, &ks[buf ^ 1][0], tid);

        int s = s0 + sl;
        if (s <= t) {
            float acc = 0.f;
            const float4* kr4 = (const float4*)&ks[buf][sl * KST];
            const float4* qr4 = (const float4*)qp_s;
            const float4* vr4 = (const float4*)vv_s;
#pragma unroll
            for (int i = 0; i < 16; ++i) {
                int h4 = part + i * 4;            // interleaved: partners read adjacent 16B
                float4 kv = kr4[h4];
                float4 qv = qr4[h4];
                float4 vw = vr4[h4];
                acc += vw.x * fast_tanh(qv.x + kv.x);
                acc += vw.y * fast_tanh(qv.y + kv.y);
                acc += vw.z * fast_tanh(qv.z + kv.z);
                acc += vw.w * fast_tanh(qv.w + kv.w);
            }
            // reduce 4 partials across adjacent lanes (partners share activity)
            acc += __shfl_xor(acc, 1);
            acc += __shfl_xor(acc, 2);
            if (part == 0) srow[s] = acc;
        }
        // Wait for the prefetched chunk, then make it (and srow) visible to all waves.
        asm volatile("s_wait_asynccnt 0x0" ::: "memory");
        __syncthreads();
    }

    // causal softmax over s = 0..t
    float lmax = -3.0e38f;
    for (int s = tid; s <= t; s += 256) lmax = fmaxf(lmax, srow[s]);
    red[tid] = lmax; __syncthreads();
    for (int off = 128; off > 0; off >>= 1) {
        if (tid < off) red[tid] = fmaxf(red[tid], red[tid + off]);
        __syncthreads();
    }
    float mx = red[0];
    __syncthreads();

    float lsum = 0.f;
    for (int s = tid; s <= t; s += 256)
        lsum += __builtin_amdgcn_exp2f((srow[s] - mx) * 1.44269504f);
    red[tid] = lsum; __syncthreads();
    for (int off = 128; off > 0; off >>= 1) {
        if (tid < off) red[tid] += red[tid + off];
        __syncthreads();
    }
    float inv = 1.0f / red[0];

    float* arow = alpha + (size_t)b * TT * TT + (size_t)t * TT;
    for (int s = tid; s < TT; s += 256)
        arow[s] = (s <= t)
                ? __builtin_amdgcn_exp2f((srow[s] - mx) * 1.44269504f) * inv
                : 0.0f;
}

// ---------------- Kernel C: ctx = alpha * K via V_WMMA_F32_16X16X4_F32
// Causal cutoff: alpha[t,s]==0 for s>t, so K-loop stops at end of query tile.
__global__ __launch_bounds__(256) void ctx_kernel(
    const float* __restrict__ alpha, const float* __restrict__ K,
    float* __restrict__ ctx)
{
    int wave = threadIdx.x >> 5;
    int lane = threadIdx.x & 31;
    int tile = blockIdx.x * 8 + wave;       // 0..1023
    int b  = tile >> 9;
    int rt = (tile & 511) >> 4;             // query tile (32)
    int ct = tile & 15;                     // h tile (16)

    int m  = lane & 15;
    int kk = (lane >> 4) * 2;

    const float* arow = alpha + (size_t)b * TT * TT + (size_t)(rt * 16 + m) * TT;
    const float* kmat = K + (size_t)b * TT * HH + ct * 16 + m;

    int sEnd = rt * 16 + 16;                // causal: zero beyond query tile
    v8f c = {};
    for (int s = 0; s < sEnd; s += 4) {
        v2f a, bm;
        a.x  = arow[s + kk];                a.y  = arow[s + kk + 1];
        bm.x = kmat[(size_t)(s + kk) * HH]; bm.y = kmat[(size_t)(s + kk + 1) * HH];
        c = __builtin_amdgcn_wmma_f32_16x16x4_f32(false, a, false, bm,
                                                  (short)0, c, false, false);
    }
    float* outBase = ctx + (size_t)b * TT * HH
                   + (size_t)(rt * 16 + 8 * (lane >> 4)) * HH + ct * 16 + m;
#pragma unroll
    for (int i = 0; i < 8; ++i) outBase[(size_t)i * HH] = c[i];
}

extern "C" void kernel_launch(void* const* d_in, const int* in_sizes, int n_in,
                              void* d_out, int out_size, void* d_ws, size_t ws_size,
                              hipStream_t stream) {
    const float* Q  = (const float*)d_in[0];
    const float* K  = (const float*)d_in[1];
    const float* Wq = (const float*)d_in[2];
    const float* Wk = (const float*)d_in[3];
    const float* v  = (const float*)d_in[4];

    float* ctx   = (float*)d_out;                          // [B,T,H]
    float* alpha = (float*)d_out + (size_t)BB * TT * HH;   // [B,T,T]

    float* Qp = (float*)d_ws;                              // [B,T,H]
    float* Kp = Qp + (size_t)BB * TT * HH;                 // [B,T,H]

    // A: projections (2048 tiles, 8 waves/block)
    proj_kernel<<<256, 256, 0, stream>>>(Q, K, Wq, Wk, Qp, Kp);
    // B: fused tanh-scores + causal softmax (one WG per (b,t))
    score_softmax_kernel<<<dim3(TT, BB), 256, 0, stream>>>(Qp, Kp, v, alpha);
    // C: ctx GEMM (1024 tiles, 8 waves/block)
    ctx_kernel<<<128, 256, 0, stream>>>(alpha, K, ctx);
}